// TransitionDown_74586402062452
// MI455X (gfx1250) — compile-verified
//
#include <hip/hip_runtime.h>

typedef __attribute__((ext_vector_type(2))) float v2f;
typedef __attribute__((ext_vector_type(8))) float v8f;

#define BB    16
#define NN    16384
#define DIN   64
#define DOUT  128
#define MM    4096
#define KK    16
#define ROWS  (BB * NN)          // 262144
#define NBLK_A (ROWS / 128)      // 2048 blocks, 8 waves x 16 rows each

// ---------------------------------------------------------------------------
// Kernel A: h = feat . W^T (bias dropped: it cancels exactly through BN).
// Each wave computes a 16x128 tile of h via V_WMMA_F32_16X16X4_F32 and
// reduces per-channel sum / sum-of-squares (deterministic LDS reduction).
// If hout != nullptr, also materializes h (Path 1).
// ---------------------------------------------------------------------------
__global__ __launch_bounds__(256) void td_gemm_stats(
    const float* __restrict__ feat, const float* __restrict__ W,
    float* __restrict__ partials, float* __restrict__ hout) {
  __shared__ float ldsW[DOUT * DIN];          // 32 KB
  __shared__ float red[8 * 2 * DOUT];         // 8 KB
  const int tid = threadIdx.x;
  for (int i = tid; i < DOUT * DIN; i += 256) ldsW[i] = W[i];
  __syncthreads();

  const int wave = tid >> 5;
  const int lane = tid & 31;
  const int lo = lane & 15;
  const int hi = lane >> 4;
  const long long r0 = ((long long)blockIdx.x * 8 + wave) * 16;
  const float* arow = feat + (r0 + lo) * DIN + 2 * hi;

  v8f zero = {};
  v8f acc[8];
#pragma unroll
  for (int t = 0; t < 8; ++t) acc[t] = zero;

  for (int kk = 0; kk < 16; ++kk) {
    v2f a = *(const v2f*)(arow + 4 * kk);
#pragma unroll
    for (int t = 0; t < 8; ++t) {
      v2f bf = *(const v2f*)(&ldsW[(16 * t + lo) * DIN + 4 * kk + 2 * hi]);
      acc[t] = __builtin_amdgcn_wmma_f32_16x16x4_f32(
          false, a, false, bf, (short)0, acc[t], false, false);
    }
  }

  // Optionally materialize h. D layout: VGPR v holds row v (lanes 0-15) and
  // row v+8 (lanes 16-31); channel = 16t + (lane & 15).
  if (hout) {
#pragma unroll
    for (int t = 0; t < 8; ++t)
#pragma unroll
      for (int v = 0; v < 8; ++v)
        hout[(r0 + v + 8 * hi) * DOUT + 16 * t + lo] = acc[t][v];
  }

  // Per-channel column sums over the 16 rows of this tile.
#pragma unroll
  for (int t = 0; t < 8; ++t) {
    float s = 0.f, q = 0.f;
#pragma unroll
    for (int v = 0; v < 8; ++v) { float e = acc[t][v]; s += e; q += e * e; }
    s += __shfl_xor(s, 16, 32);
    q += __shfl_xor(q, 16, 32);
    if (lane < 16) {
      red[wave * 256 + 16 * t + lane] = s;
      red[wave * 256 + DOUT + 16 * t + lane] = q;
    }
  }
  __syncthreads();

  float a = 0.f;
#pragma unroll
  for (int w = 0; w < 8; ++w) a += red[w * 256 + tid];
  partials[(long long)blockIdx.x * 256 + tid] = a;
}

// ---------------------------------------------------------------------------
// Kernel B: finalize BN stats in f64, emit fused affine: scale/shift.
// ---------------------------------------------------------------------------
__global__ void td_stats_finalize(const float* __restrict__ partials,
                                  const float* __restrict__ gamma,
                                  const float* __restrict__ beta,
                                  float* __restrict__ stats) {
  const int o = threadIdx.x;
  if (o >= DOUT) return;
  double s = 0.0, q = 0.0;
  for (int p = 0; p < NBLK_A; ++p) {
    s += (double)partials[p * 256 + o];
    q += (double)partials[p * 256 + DOUT + o];
  }
  const double inv = 1.0 / (double)ROWS;
  const double mu = s * inv;
  const double var = q * inv - mu * mu;
  const float scale = gamma[o] * (float)(1.0 / sqrt(var + 1e-5));
  const float shift = beta[o] - (float)mu * scale;
  stats[o] = scale;
  stats[DOUT + o] = shift;
}

// ---------------------------------------------------------------------------
// Kernel C (Path 1): gather rows of the materialized h (L2-resident: 128 MB
// fits the 192 MB L2), apply the fused BN affine + ReLU(max over K). One
// wave per (b,m); each lane owns 4 channels -> one b128 load per row.
// ---------------------------------------------------------------------------
__global__ __launch_bounds__(256) void td_gather_pre(
    const float* __restrict__ pos, const float* __restrict__ h,
    const int* __restrict__ fps, const int* __restrict__ kidx,
    const float* __restrict__ stats, float* __restrict__ out_pos,
    float* __restrict__ out_feat) {
  const int tid = threadIdx.x;
  const int wave = tid >> 5;
  const int lane = tid & 31;
  const int g = blockIdx.x * 8 + wave;        // 0 .. B*M-1
  const int b = g >> 12;                      // g / M

  const int nb = (lane < KK) ? kidx[g * KK + lane] : 0;
  const float4 sc = *(const float4*)(stats + lane * 4);
  const float4 sh = *(const float4*)(stats + DOUT + lane * 4);

  float4 m4;
  m4.x = m4.y = m4.z = m4.w = -3.402823466e38f;
#pragma unroll
  for (int k = 0; k < KK; ++k) {
    const int row = __shfl(nb, k, 32);
    const float4 hv = *(const float4*)(h + ((long long)b * NN + row) * DOUT +
                                       lane * 4);
    m4.x = fmaxf(m4.x, fmaf(hv.x, sc.x, sh.x));
    m4.y = fmaxf(m4.y, fmaf(hv.y, sc.y, sh.y));
    m4.z = fmaxf(m4.z, fmaf(hv.z, sc.z, sh.z));
    m4.w = fmaxf(m4.w, fmaf(hv.w, sc.w, sh.w));
  }
  m4.x = fmaxf(m4.x, 0.f); m4.y = fmaxf(m4.y, 0.f);
  m4.z = fmaxf(m4.z, 0.f); m4.w = fmaxf(m4.w, 0.f);
  *(float4*)(out_feat + (long long)g * DOUT + lane * 4) = m4;

  if (lane < 3) {
    const int fi = fps[g];
    out_pos[(long long)g * 3 + lane] = pos[((long long)b * NN + fi) * 3 + lane];
  }
}

// ---------------------------------------------------------------------------
// Kernel C (Path 2, small workspace): recompute h on the fly. Per (b,m):
// gather K=16 neighbor feat rows as a 16x64 WMMA A-matrix, multiply by W,
// apply BN affine, ReLU(max over the 16 rows == tile M dimension).
// ---------------------------------------------------------------------------
__global__ __launch_bounds__(256) void td_gather_fused(
    const float* __restrict__ pos, const float* __restrict__ feat,
    const int* __restrict__ fps, const int* __restrict__ kidx,
    const float* __restrict__ W, const float* __restrict__ stats,
    float* __restrict__ out_pos, float* __restrict__ out_feat) {
  __shared__ float ldsW[DOUT * DIN];
  __shared__ float ldsS[2 * DOUT];
  const int tid = threadIdx.x;
  for (int i = tid; i < DOUT * DIN; i += 256) ldsW[i] = W[i];
  if (tid < 2 * DOUT) ldsS[tid] = stats[tid];
  __syncthreads();

  const int wave = tid >> 5;
  const int lane = tid & 31;
  const int lo = lane & 15;
  const int hi = lane >> 4;
  const int g = blockIdx.x * 8 + wave;
  const int b = g >> 12;

  const long long nb = (long long)kidx[g * KK + lo];  // all lanes active
  const float* arow = feat + ((long long)b * NN + nb) * DIN + 2 * hi;

  v8f zero = {};
  v8f acc[8];
#pragma unroll
  for (int t = 0; t < 8; ++t) acc[t] = zero;

  for (int kk = 0; kk < 16; ++kk) {
    v2f a = *(const v2f*)(arow + 4 * kk);
#pragma unroll
    for (int t = 0; t < 8; ++t) {
      v2f bf = *(const v2f*)(&ldsW[(16 * t + lo) * DIN + 4 * kk + 2 * hi]);
      acc[t] = __builtin_amdgcn_wmma_f32_16x16x4_f32(
          false, a, false, bf, (short)0, acc[t], false, false);
    }
  }

#pragma unroll
  for (int t = 0; t < 8; ++t) {
    const float sc = ldsS[16 * t + lo];
    const float sh = ldsS[DOUT + 16 * t + lo];
    float mx = -3.402823466e38f;
#pragma unroll
    for (int v = 0; v < 8; ++v) mx = fmaxf(mx, fmaf(acc[t][v], sc, sh));
    mx = fmaxf(mx, __shfl_xor(mx, 16, 32));   // combine rows 0-7 with 8-15
    mx = fmaxf(mx, 0.0f);                     // ReLU (monotone, safe post-max)
    if (lane < 16) out_feat[(long long)g * DOUT + 16 * t + lane] = mx;
  }

  if (lane < 3) {
    const int fi = fps[g];
    out_pos[(long long)g * 3 + lane] = pos[((long long)b * NN + fi) * 3 + lane];
  }
}

// ---------------------------------------------------------------------------
extern "C" void kernel_launch(void* const* d_in, const int* in_sizes, int n_in,
                              void* d_out, int out_size, void* d_ws,
                              size_t ws_size, hipStream_t stream) {
  (void)in_sizes; (void)n_in; (void)out_size;
  const float* pos   = (const float*)d_in[0];
  const float* feat  = (const float*)d_in[1];
  const int*   fps   = (const int*)d_in[2];
  const int*   kidx  = (const int*)d_in[3];
  const float* W     = (const float*)d_in[4];
  // d_in[5] = bias: cancels exactly through BatchNorm, unused.
  const float* gamma = (const float*)d_in[6];
  const float* beta  = (const float*)d_in[7];

  float* out_pos  = (float*)d_out;                    // [B, M, 3]
  float* out_feat = out_pos + (size_t)BB * MM * 3;    // [B, M, 128]

  const size_t h_elems    = (size_t)ROWS * DOUT;          // 33.5 M floats
  const size_t part_elems = (size_t)NBLK_A * 256;         // 0.5 M floats
  const size_t need_big   = (h_elems + part_elems + 256) * sizeof(float);

  if (ws_size >= need_big) {
    // Path 1: materialize h (fits L2), gather becomes pure memory op.
    float* h        = (float*)d_ws;
    float* partials = h + h_elems;
    float* stats    = partials + part_elems;
    td_gemm_stats<<<NBLK_A, 256, 0, stream>>>(feat, W, partials, h);
    td_stats_finalize<<<1, 128, 0, stream>>>(partials, gamma, beta, stats);
    td_gather_pre<<<(BB * MM) / 8, 256, 0, stream>>>(pos, h, fps, kidx, stats,
                                                     out_pos, out_feat);
  } else {
    // Path 2: tiny workspace; recompute h in the gather via WMMA.
    float* partials = (float*)d_ws;
    float* stats    = partials + part_elems;
    td_gemm_stats<<<NBLK_A, 256, 0, stream>>>(feat, W, partials, nullptr);
    td_stats_finalize<<<1, 128, 0, stream>>>(partials, gamma, beta, stats);
    td_gather_fused<<<(BB * MM) / 8, 256, 0, stream>>>(
        pos, feat, fps, kidx, W, stats, out_pos, out_feat);
  }
}